// QuantumDARTSModel_88235808129069
// MI455X (gfx1250) — compile-verified
//
#include <hip/hip_runtime.h>
#include <math.h>

// ---------------------------------------------------------------------------
// QuantumDARTS forward on MI455X (gfx1250, wave32).
//
// Math: the 8x16 grid of single-qubit gates collapses (tensor-product algebra)
// to ONE tensor product of 16 per-qubit 2x2 complex matrices, grouped into
// four 16x16 complex matrices G0..G3 (G0 acts on state-index bits 15:12, ...,
// G3 on bits 3:0).  State update = two passes of 16x16 complex tile matmuls,
// implemented with chained V_WMMA_F32_16X16X4_F32 (fp32, matches reference).
// Bandwidth-bound: state re+im = 128 MB fp32, fits in 192 MB L2.
// ---------------------------------------------------------------------------

#define NQ 16
#define NL 8
#define NB 256
#define NS 65536  // 2^16 amplitudes

typedef __attribute__((ext_vector_type(2))) float v2f;
typedef __attribute__((ext_vector_type(8))) float v8f;

// D(16x16,f32) += A(16x4,f32) x B(4x16,f32)   — VOP3P V_WMMA_F32_16X16X4_F32
__device__ __forceinline__ v8f wmma4(v2f a, v2f b, v8f c) {
  return __builtin_amdgcn_wmma_f32_16x16x4_f32(
      /*neg_a=*/false, a, /*neg_b=*/false, b,
      /*c_mod=*/(short)0, c, /*reuse_a=*/false, /*reuse_b=*/false);
}

// A-layout slice k (= columns 4k..4k+3) of a row-major 16x16 matrix g.
// ISA: A 16x4 f32: lanes 0-15 -> {K=0,1}, lanes 16-31 -> {K=2,3}, M = lane%16.
__device__ __forceinline__ v2f aslice(const float* __restrict__ g, int k, int lane) {
  int m = lane & 15;
  int c = 4 * k + ((lane >> 4) << 1);  // +0 (lanes<16) or +2 (lanes>=16)
  v2f a;
  a[0] = g[m * 16 + c];
  a[1] = g[m * 16 + c + 1];
  return a;
}

// B-layout slice k of X^T where X is a row-major 16x16 tile in LDS:
// B[kk][n] = X^T[4k+kk][n] = X[n][4k+kk].
// B 4x16 f32 (mirrors A/C split): lanes 0-15 -> rows {4k,4k+1}, lanes 16-31 -> {4k+2,4k+3}.
__device__ __forceinline__ v2f bsliceT(const float* x, int k, int lane) {
  int n = lane & 15;
  int r = 4 * k + ((lane >> 4) << 1);
  v2f b;
  b[0] = x[n * 16 + r];
  b[1] = x[n * 16 + r + 1];
  return b;
}

// C/D layout: VGPR r holds D[r][lane] (lanes 0-15) and D[r+8][lane-16] (lanes 16-31).
__device__ __forceinline__ void storeC_lds(float* dst, v8f d, int lane) {
  int col = lane & 15;
  int row0 = (lane >> 4) << 3;
#pragma unroll
  for (int r = 0; r < 8; ++r) dst[(row0 + r) * 16 + col] = d[r];
}

__device__ __forceinline__ float hash_u01(unsigned v) {
  v ^= 0x9e3779b9u;
  v ^= v >> 17; v *= 0xed5ad4bbu;
  v ^= v >> 11; v *= 0xac4c1b51u;
  v ^= v >> 15; v *= 0x31848babu;
  v ^= v >> 14;
  return ((float)(v >> 8) + 0.5f) * (1.0f / 16777216.0f);
}

// ---------------------------------------------------------------------------
// Kernel 1: gate selection (argmax of logits+gumbel), per-qubit layer product,
// group tensor products -> Gr/Gi [4][16][16] in workspace.
// ---------------------------------------------------------------------------
__global__ void prep_kernel(const float* __restrict__ P, const float* __restrict__ Q,
                            const float* __restrict__ rot,
                            float* __restrict__ gr, float* __restrict__ gi) {
  __shared__ float mr[NL * NQ][2][2], mi[NL * NQ][2][2];
  __shared__ float pr[NQ][2][2], pi[NQ][2][2];
  int tid = threadIdx.x;

  if (tid < NL * NQ) {
    const float* p = P + tid * 4;   // [1,4] row
    const float* qm = Q + tid * 16; // [4,4]
    float best = -1e30f;
    int bi = 0;
#pragma unroll
    for (int g = 0; g < 4; ++g) {
      float lg = 0.f;
#pragma unroll
      for (int k = 0; k < 4; ++k) lg += p[k] * qm[k * 4 + g];
      float u = hash_u01((unsigned)(tid * 4 + g));
      float gum = -__logf(-__logf(u));
      float sc = lg + gum;  // tau=1; hard one-hot forward == argmax
      if (sc > best) { best = sc; bi = g; }
    }
    float half = rot[tid] * 0.5f;
    float c = __cosf(half), s = __sinf(half);
    float hh = 0.70710678118654752f;
    float ur[2][2], ui[2][2];
    if (bi == 0) {        // rx
      ur[0][0] = c; ur[0][1] = 0; ur[1][0] = 0; ur[1][1] = c;
      ui[0][0] = 0; ui[0][1] = -s; ui[1][0] = -s; ui[1][1] = 0;
    } else if (bi == 1) { // ry
      ur[0][0] = c; ur[0][1] = -s; ur[1][0] = s; ur[1][1] = c;
      ui[0][0] = 0; ui[0][1] = 0; ui[1][0] = 0; ui[1][1] = 0;
    } else if (bi == 2) { // rz
      ur[0][0] = c; ur[0][1] = 0; ur[1][0] = 0; ur[1][1] = c;
      ui[0][0] = -s; ui[0][1] = 0; ui[1][0] = 0; ui[1][1] = s;
    } else {              // h
      ur[0][0] = hh; ur[0][1] = hh; ur[1][0] = hh; ur[1][1] = -hh;
      ui[0][0] = 0; ui[0][1] = 0; ui[1][0] = 0; ui[1][1] = 0;
    }
#pragma unroll
    for (int i = 0; i < 2; ++i)
#pragma unroll
      for (int j = 0; j < 2; ++j) { mr[tid][i][j] = ur[i][j]; mi[tid][i][j] = ui[i][j]; }
  }
  __syncthreads();

  if (tid < NQ) {
    // G_q = M_{7,q} * ... * M_{0,q}  (layer 0 applied first)
    float g2r[2][2] = {{1, 0}, {0, 1}}, g2i[2][2] = {{0, 0}, {0, 0}};
    for (int l = 0; l < NL; ++l) {
      int id = l * NQ + tid;
      float nr[2][2], ni[2][2];
#pragma unroll
      for (int i = 0; i < 2; ++i)
#pragma unroll
        for (int j = 0; j < 2; ++j) {
          float ar = 0.f, ai = 0.f;
#pragma unroll
          for (int k = 0; k < 2; ++k) {
            ar += mr[id][i][k] * g2r[k][j] - mi[id][i][k] * g2i[k][j];
            ai += mr[id][i][k] * g2i[k][j] + mi[id][i][k] * g2r[k][j];
          }
          nr[i][j] = ar; ni[i][j] = ai;
        }
#pragma unroll
      for (int i = 0; i < 2; ++i)
#pragma unroll
        for (int j = 0; j < 2; ++j) { g2r[i][j] = nr[i][j]; g2i[i][j] = ni[i][j]; }
    }
#pragma unroll
    for (int i = 0; i < 2; ++i)
#pragma unroll
      for (int j = 0; j < 2; ++j) { pr[tid][i][j] = g2r[i][j]; pi[tid][i][j] = g2i[i][j]; }
  }
  __syncthreads();

  // Group k = qubits 4k..4k+3; qubit 4k is the MSB of the nibble.
  for (int t = 0; t < 4; ++t) {
    int idx = tid + t * 256;          // 0..1023
    int k = idx >> 8, rem = idx & 255;
    int i = rem >> 4, j = rem & 15;
    float ar = 1.f, ai = 0.f;
#pragma unroll
    for (int tq = 0; tq < 4; ++tq) {
      int q = 4 * k + tq;
      int bi_ = (i >> (3 - tq)) & 1, bj_ = (j >> (3 - tq)) & 1;
      float fr = pr[q][bi_][bj_], fi = pi[q][bi_][bj_];
      float nr = ar * fr - ai * fi, ni = ar * fi + ai * fr;
      ar = nr; ai = ni;
    }
    gr[idx] = ar;
    gi[idx] = ai;
  }
}

// ---------------------------------------------------------------------------
// Kernel 2: per-row inverse norms.
// ---------------------------------------------------------------------------
__global__ void norm_kernel(const float* __restrict__ x, float* __restrict__ inv_norm) {
  __shared__ float red[256];
  int b = blockIdx.x, tid = threadIdx.x;
  float s = 0.f;
  for (int i = tid; i < NS; i += 256) {
    float v = x[(size_t)b * NS + i];
    s += v * v;
  }
  red[tid] = s;
  __syncthreads();
  for (int o = 128; o; o >>= 1) {
    if (tid < o) red[tid] += red[tid + o];
    __syncthreads();
  }
  if (tid == 0) inv_norm[b] = 1.0f / (sqrtf(red[0]) + 1e-12f);
}

// ---------------------------------------------------------------------------
// Kernel 3 (pass 1): contiguous tiles S[i1][i0]; S' = G2 * S * G3^T, S real.
// One wave per tile; 24 WMMAs per tile. 8192 blocks x 8 waves = 65536 tiles.
// ---------------------------------------------------------------------------
__global__ void pass1_kernel(const float* __restrict__ x, const float* __restrict__ inv_norm,
                             const float* __restrict__ gr, const float* __restrict__ gi,
                             float* __restrict__ sre, float* __restrict__ sim) {
  __shared__ float lS[8][256], lU1[8][256], lU2[8][256];  // 24 KB
  int lane = threadIdx.x & 31;
  int w = threadIdx.x >> 5;
  int t = blockIdx.x * 8 + w;  // 0..65535
  int b = t >> 8, tt = t & 255;
  int base = b * NS + tt * 256;
  float inv = inv_norm[b];
  float* S = lS[w];
  float* U1 = lU1[w];
  float* U2 = lU2[w];

#pragma unroll
  for (int i = 0; i < 8; ++i) S[lane + i * 32] = x[base + lane + i * 32] * inv;

  v2f a3r[4], a3i[4], a2r[4], a2i[4], a2in[4];
#pragma unroll
  for (int k = 0; k < 4; ++k) {
    a3r[k] = aslice(gr + 3 * 256, k, lane);
    a3i[k] = aslice(gi + 3 * 256, k, lane);
    a2r[k] = aslice(gr + 2 * 256, k, lane);
    a2i[k] = aslice(gi + 2 * 256, k, lane);
    a2in[k][0] = -a2i[k][0];
    a2in[k][1] = -a2i[k][1];
  }
  __syncthreads();

  // U = (S * G3^T)^T = G3 * S^T ;  S real -> Ur = G3r S^T, Ui = G3i S^T
  v8f u1 = {}, u2 = {};
#pragma unroll
  for (int k = 0; k < 4; ++k) {
    v2f bs = bsliceT(S, k, lane);
    u1 = wmma4(a3r[k], bs, u1);
    u2 = wmma4(a3i[k], bs, u2);
  }
  storeC_lds(U1, u1, lane);
  storeC_lds(U2, u2, lane);
  __syncthreads();

  // S' = G2 * U^T :  S'r = G2r Ur^T - G2i Ui^T ; S'i = G2r Ui^T + G2i Ur^T
  v8f dr = {}, di = {};
#pragma unroll
  for (int k = 0; k < 4; ++k) {
    v2f b1 = bsliceT(U1, k, lane);
    v2f b2 = bsliceT(U2, k, lane);
    dr = wmma4(a2r[k], b1, dr);
    dr = wmma4(a2in[k], b2, dr);
    di = wmma4(a2r[k], b2, di);
    di = wmma4(a2i[k], b1, di);
  }
  int col = lane & 15, row0 = (lane >> 4) << 3;
#pragma unroll
  for (int r = 0; r < 8; ++r) {
    sre[base + (row0 + r) * 16 + col] = dr[r];
    sim[base + (row0 + r) * 16 + col] = di[r];
  }
}

// ---------------------------------------------------------------------------
// Kernel 4 (pass 2, in-place): strided tiles S[i3][i2]; S'' = G0 * S * G1^T,
// S complex. 32 WMMAs per tile.
// ---------------------------------------------------------------------------
__global__ void pass2_kernel(float* __restrict__ sre, float* __restrict__ sim,
                             const float* __restrict__ gr, const float* __restrict__ gi) {
  __shared__ float lSr[8][256], lSi[8][256], lU1[8][256], lU2[8][256];  // 32 KB
  int lane = threadIdx.x & 31;
  int w = threadIdx.x >> 5;
  int t = blockIdx.x * 8 + w;
  int b = t >> 8, low = t & 255;
  int base = b * NS + low;
  float* Sr = lSr[w];
  float* Si = lSi[w];
  float* U1 = lU1[w];
  float* U2 = lU2[w];

#pragma unroll
  for (int i = 0; i < 8; ++i) {
    int idx = lane + i * 32;                              // i3*16+i2
    int off = base + (idx >> 4) * 4096 + (idx & 15) * 256;
    Sr[idx] = sre[off];
    Si[idx] = sim[off];
  }

  v2f a1r[4], a1i[4], a1in[4], a0r[4], a0i[4], a0in[4];
#pragma unroll
  for (int k = 0; k < 4; ++k) {
    a1r[k] = aslice(gr + 1 * 256, k, lane);
    a1i[k] = aslice(gi + 1 * 256, k, lane);
    a0r[k] = aslice(gr, k, lane);
    a0i[k] = aslice(gi, k, lane);
    a1in[k][0] = -a1i[k][0]; a1in[k][1] = -a1i[k][1];
    a0in[k][0] = -a0i[k][0]; a0in[k][1] = -a0i[k][1];
  }
  __syncthreads();

  // U = G1 * S^T (complex): Ur = G1r Sr^T - G1i Si^T ; Ui = G1r Si^T + G1i Sr^T
  v8f u1 = {}, u2 = {};
#pragma unroll
  for (int k = 0; k < 4; ++k) {
    v2f br = bsliceT(Sr, k, lane);
    v2f bi2 = bsliceT(Si, k, lane);
    u1 = wmma4(a1r[k], br, u1);
    u1 = wmma4(a1in[k], bi2, u1);
    u2 = wmma4(a1r[k], bi2, u2);
    u2 = wmma4(a1i[k], br, u2);
  }
  storeC_lds(U1, u1, lane);
  storeC_lds(U2, u2, lane);
  __syncthreads();

  // S'' = G0 * U^T (complex)
  v8f dr = {}, di = {};
#pragma unroll
  for (int k = 0; k < 4; ++k) {
    v2f b1 = bsliceT(U1, k, lane);
    v2f b2 = bsliceT(U2, k, lane);
    dr = wmma4(a0r[k], b1, dr);
    dr = wmma4(a0in[k], b2, dr);
    di = wmma4(a0r[k], b2, di);
    di = wmma4(a0i[k], b1, di);
  }
  int col = lane & 15, row0 = (lane >> 4) << 3;
#pragma unroll
  for (int r = 0; r < 8; ++r) {
    int off = base + (row0 + r) * 4096 + col * 256;
    sre[off] = dr[r];
    sim[off] = di[r];
  }
}

// ---------------------------------------------------------------------------
// Kernel 5: class marginals (top 2 index bits = class), normalized.
// ---------------------------------------------------------------------------
__global__ void marg_kernel(const float* __restrict__ sre, const float* __restrict__ sim,
                            float* __restrict__ out) {
  __shared__ float red[256];
  __shared__ float cls[4];
  int b = blockIdx.x, tid = threadIdx.x;
  for (int c = 0; c < 4; ++c) {
    float s = 0.f;
    for (int i = c * 64; i < c * 64 + 64; ++i) {
      int off = b * NS + i * 256 + tid;
      float re = sre[off], im = sim[off];
      s += re * re + im * im;
    }
    red[tid] = s;
    __syncthreads();
    for (int o = 128; o; o >>= 1) {
      if (tid < o) red[tid] += red[tid + o];
      __syncthreads();
    }
    if (tid == 0) cls[c] = red[0];
    __syncthreads();
  }
  if (tid < 4) {
    float tot = cls[0] + cls[1] + cls[2] + cls[3];
    out[b * 4 + tid] = cls[tid] / (tot + 1e-12f);
  }
}

// ---------------------------------------------------------------------------
extern "C" void kernel_launch(void* const* d_in, const int* in_sizes, int n_in,
                              void* d_out, int out_size, void* d_ws, size_t ws_size,
                              hipStream_t stream) {
  const float* x = (const float*)d_in[0];    // [256, 65536]
  const float* P = (const float*)d_in[1];    // [8,16,1,4]
  const float* Q = (const float*)d_in[2];    // [8,16,4,4]
  const float* rot = (const float*)d_in[3];  // [128]
  float* out = (float*)d_out;                // [256, 4]

  float* ws = (float*)d_ws;
  float* gr = ws;                 // 4*256 group matrices, real
  float* gi = ws + 1024;          // imag
  float* inv_norm = ws + 2048;    // 256
  float* sre = ws + 4096;         // 16M state real
  float* sim = sre + (size_t)NB * NS;  // 16M state imag

  prep_kernel<<<1, 256, 0, stream>>>(P, Q, rot, gr, gi);
  norm_kernel<<<NB, 256, 0, stream>>>(x, inv_norm);
  pass1_kernel<<<(NB * 256) / 8, 256, 0, stream>>>(x, inv_norm, gr, gi, sre, sim);
  pass2_kernel<<<(NB * 256) / 8, 256, 0, stream>>>(sre, sim, gr, gi);
  marg_kernel<<<NB, 256, 0, stream>>>(sre, sim, out);
}